// PFNLayerV2_61366492725933
// MI455X (gfx1250) — compile-verified
//
#include <hip/hip_runtime.h>
#include <hip/hip_bf16.h>

#define IN_CH  10
#define OUT_CH 32
#define BN_EPS 1e-3f

typedef __attribute__((ext_vector_type(2))) float v2f;
typedef __attribute__((ext_vector_type(8))) float v8f;

// ---------------------------------------------------------------------------
// Kernel 1: zero the pillar-max workspace (count read from device scalar so
// the launch shape is capture-safe / deterministic).
// ---------------------------------------------------------------------------
__global__ void pfn_zero_ws(float* __restrict__ ws, const int* __restrict__ num_pillars_p) {
    const int total = num_pillars_p[0] * OUT_CH;
    for (int i = blockIdx.x * blockDim.x + threadIdx.x; i < total;
         i += gridDim.x * blockDim.x)
        ws[i] = 0.0f;
}

// ---------------------------------------------------------------------------
// Per-tile body: 16 points x 32 channels via two 16x16 fp32 WMMA tiles,
// 3 K-steps of 16x16x4 (K padded 10 -> 12). FULL=true: no bounds checks
// (uniform fast path); FULL=false: per-row guards for the tail tile.
// ---------------------------------------------------------------------------
template <bool FULL>
__device__ __forceinline__ void pfn_tile(
    const float* __restrict__ in, const int* __restrict__ unq_inv,
    float* __restrict__ out, float* __restrict__ ws,
    const v2f (&bfrag)[2][3],
    float s0, float o0, float s1, float o1,
    int row0, int lm, int half, int n0, int n1, int n)
{
    // ---- A fragments: 16x4 f32 tile per K-step.
    // lanes 0-15: M=lane, K=k0..k0+1 ; lanes 16-31: M=lane-16, K=k0+2..k0+3
    const int  rowA   = row0 + lm;
    const bool rvalid = FULL || (rowA < n);
    const int  rowC   = rvalid ? rowA : (n - 1);
    v2f afrag[3];
#pragma unroll
    for (int ks = 0; ks < 3; ++ks) {
        const int k = ks * 4 + half * 2;
        v2f a;
        if (FULL) {
            a.x = (k     < IN_CH) ? in[(size_t)rowA * IN_CH + k]     : 0.0f;
            a.y = (k + 1 < IN_CH) ? in[(size_t)rowA * IN_CH + k + 1] : 0.0f;
        } else {
            a.x = (rvalid && k     < IN_CH) ? in[(size_t)rowC * IN_CH + k]     : 0.0f;
            a.y = (rvalid && k + 1 < IN_CH) ? in[(size_t)rowC * IN_CH + k + 1] : 0.0f;
        }
        afrag[ks] = a;
    }

    // ---- D = A x B (+C), fp32 WMMA, two N-tiles
    v8f c0 = {};
    v8f c1 = {};
#pragma unroll
    for (int ks = 0; ks < 3; ++ks) {
        c0 = __builtin_amdgcn_wmma_f32_16x16x4_f32(
            false, afrag[ks], false, bfrag[0][ks], (short)0, c0, false, false);
        c1 = __builtin_amdgcn_wmma_f32_16x16x4_f32(
            false, afrag[ks], false, bfrag[1][ks], (short)0, c1, false, false);
    }

    // ---- BN + ReLU + store + run-length-compacted atomic scatter-max.
    // D layout: VGPR r holds row M = r + 8*half, col N = n0 (tile0) / n1 (tile1).
    // unq_inv is sorted -> consecutive rows in a lane usually share a pillar.
    int   curP = -1;
    float m0 = 0.0f, m1 = 0.0f;
#pragma unroll
    for (int r = 0; r < 8; ++r) {
        const int row = row0 + r + half * 8;
        if (FULL || row < n) {
            const float x0 = fmaxf(c0[r] * s0 + o0, 0.0f);
            const float x1 = fmaxf(c1[r] * s1 + o1, 0.0f);
            out[(size_t)row * 64 + n0] = x0;
            out[(size_t)row * 64 + n1] = x1;
            const int p = unq_inv[row];
            if (p != curP) {
                if (curP >= 0) {
                    atomicMax((unsigned int*)&ws[(size_t)curP * OUT_CH + n0],
                              __float_as_uint(m0));
                    atomicMax((unsigned int*)&ws[(size_t)curP * OUT_CH + n1],
                              __float_as_uint(m1));
                }
                curP = p; m0 = x0; m1 = x1;
            } else {
                m0 = fmaxf(m0, x0);
                m1 = fmaxf(m1, x1);
            }
        }
    }
    if (curP >= 0) {
        atomicMax((unsigned int*)&ws[(size_t)curP * OUT_CH + n0], __float_as_uint(m0));
        atomicMax((unsigned int*)&ws[(size_t)curP * OUT_CH + n1], __float_as_uint(m1));
    }
}

// ---------------------------------------------------------------------------
// Kernel 2: fp32 WMMA GEMM + fused BN(eval) + ReLU + scatter-max.
// One wave = 16 points; 8 waves per 256-thread block.
// ---------------------------------------------------------------------------
__global__ __launch_bounds__(256) void pfn_gemm_bn_relu_scatter(
    const float* __restrict__ in,       // [N,10]
    const int*   __restrict__ unq_inv,  // [N] (sorted)
    const float* __restrict__ W,        // [32,10] row-major
    const float* __restrict__ gamma,
    const float* __restrict__ beta,
    const float* __restrict__ rmean,
    const float* __restrict__ rvar,
    float*       __restrict__ out,      // [N,64]
    float*       __restrict__ ws,       // [P,32]
    int n)
{
    const int lane = threadIdx.x & 31;
    const int wave = threadIdx.x >> 5;
    const int row0 = (blockIdx.x * 8 + wave) * 16;   // uniform per wave
    if (row0 >= n) return;                           // whole-wave exit: EXEC stays full
    const int half = lane >> 4;                      // 0: lanes 0-15, 1: lanes 16-31
    const int lm   = lane & 15;

    // ---- B fragments: B[k][nn] = W[nn][k]; 2 N-tiles x 3 K-steps, loaded once.
    // VGPR0(.x): K = k0 + 2*half ; VGPR1(.y): K = k0 + 2*half + 1 (mirrors A layout)
    v2f bfrag[2][3];
#pragma unroll
    for (int t = 0; t < 2; ++t) {
        const int nn = t * 16 + lm;
#pragma unroll
        for (int ks = 0; ks < 3; ++ks) {
            const int k = ks * 4 + half * 2;
            v2f b;
            b.x = (k     < IN_CH) ? W[nn * IN_CH + k]     : 0.0f;
            b.y = (k + 1 < IN_CH) ? W[nn * IN_CH + k + 1] : 0.0f;
            bfrag[t][ks] = b;
        }
    }

    // ---- BN constants for this lane's two columns
    const int n0 = lm, n1 = lm + 16;
    const float s0 = gamma[n0] * rsqrtf(rvar[n0] + BN_EPS);
    const float o0 = beta[n0] - rmean[n0] * s0;
    const float s1 = gamma[n1] * rsqrtf(rvar[n1] + BN_EPS);
    const float o1 = beta[n1] - rmean[n1] * s1;

    // Wave-uniform branch: fast path has zero per-element exec-mask churn.
    if (row0 + 16 <= n) {
        pfn_tile<true >(in, unq_inv, out, ws, bfrag, s0, o0, s1, o1,
                        row0, lm, half, n0, n1, n);
    } else {
        pfn_tile<false>(in, unq_inv, out, ws, bfrag, s0, o0, s1, o1,
                        row0, lm, half, n0, n1, n);
    }
}

// ---------------------------------------------------------------------------
// Kernel 3: gather pillar maxima back per point -> out[:, 32:64].
// 32 consecutive threads share one row: coalesced 128B reads/writes,
// pillar buffer lives in L2 (3.84 MB << 192 MB).
// ---------------------------------------------------------------------------
__global__ void pfn_gather(const int* __restrict__ unq_inv,
                           const float* __restrict__ ws,
                           float* __restrict__ out, int n)
{
    const long long idx   = (long long)blockIdx.x * blockDim.x + threadIdx.x;
    const long long total = (long long)n * OUT_CH;
    if (idx >= total) return;
    const int row = (int)(idx >> 5);
    const int ch  = (int)(idx & 31);
    const int p   = unq_inv[row];
    out[(size_t)row * 64 + OUT_CH + ch] = ws[(size_t)p * OUT_CH + ch];
}

// ---------------------------------------------------------------------------
extern "C" void kernel_launch(void* const* d_in, const int* in_sizes, int n_in,
                              void* d_out, int out_size, void* d_ws, size_t ws_size,
                              hipStream_t stream) {
    const float* in      = (const float*)d_in[0];
    const int*   unq_inv = (const int*)  d_in[1];
    const int*   nump    = (const int*)  d_in[2];   // scalar num_out_inds
    const float* W       = (const float*)d_in[3];
    const float* gamma   = (const float*)d_in[4];
    const float* beta    = (const float*)d_in[5];
    const float* rmean   = (const float*)d_in[6];
    const float* rvar    = (const float*)d_in[7];
    float* out = (float*)d_out;
    float* ws  = (float*)d_ws;

    const int n = in_sizes[0] / IN_CH;   // number of points

    // 1) zero pillar-max buffer (grid-stride; element count read on device)
    pfn_zero_ws<<<1024, 256, 0, stream>>>(ws, nump);

    // 2) WMMA GEMM + BN + ReLU + scatter-max; one wave per 16 points
    const int tiles  = (n + 15) / 16;
    const int blocks = (tiles + 7) / 8;      // 8 waves (256 threads) per block
    pfn_gemm_bn_relu_scatter<<<blocks, 256, 0, stream>>>(
        in, unq_inv, W, gamma, beta, rmean, rvar, out, ws, n);

    // 3) gather maxima into out[:, 32:64]
    const long long total  = (long long)n * OUT_CH;
    const int gblocks = (int)((total + 255) / 256);
    pfn_gather<<<gblocks, 256, 0, stream>>>(unq_inv, ws, out, n);
}